// LSTM_70119636074973
// MI455X (gfx1250) — compile-verified
//
#include <hip/hip_runtime.h>
#include <hip/hip_bf16.h>
#include <math.h>

// ---------------------------------------------------------------------------
// LSTM-ish scan + vocab projection for MI455X (gfx1250, wave32, WMMA)
//
// Phase 1 (parallel GEMM, WMMA f32 16x16x4):
//   Zx[m, n] = b[n] + sum_k embed[tokens[m]][k] * W[n][k]      (k < 1024)
// Phase 2 (sequential scan, 4 WGs = one per batch):
//   z = Zx[t] + h @ Wh.T ; LayerNorm ; gelu/tanh ; c,h update
// Phase 3 (parallel GEMM, WMMA f32 16x16x4):
//   logits[m, v] = b_cls[v] + sum_k Hout[m][k] * embed[v][k]
//
// GEMM tiling: block = 256 thr (8 waves), block tile 128(M) x 256(N),
// wave tile 64x64 = 4x4 WMMA tiles -> 16 WMMA : 8 b64-loads per k-step
// (16 flop/byte wave-local, ~44 flop/byte block-level reuse).
// ---------------------------------------------------------------------------

typedef __attribute__((ext_vector_type(2))) float v2f;
typedef __attribute__((ext_vector_type(8))) float v8f;

#define E_DIM   1024
#define S_LEN   512
#define B_SZ    4
#define V_SZ    32000
#define M_TOT   (B_SZ * S_LEN)   // 2048

static __device__ __forceinline__ v8f wmma_f32(v2f a, v2f b, v8f c) {
    return __builtin_amdgcn_wmma_f32_16x16x4_f32(
        /*neg_a=*/false, a, /*neg_b=*/false, b,
        /*c_mod=*/(short)0, c, /*reuse_a=*/false, /*reuse_b=*/false);
}

// ---------------------------------------------------------------------------
// 64x64-per-wave fp32 WMMA GEMM body.
//   C[m, n] = bias[n] + sum_{k<1024} Arow[i](k) * Brow[j](k)
// Arow[i]/Bb[j] point at the start of row (m0+i*16+rr)/(n0+j*16+rr), with
// contiguous K. A-frag lane map (32-bit A, 16x4): lane<16 -> K={0,1},
// lane>=16 -> K={2,3}, row = lane&15 -> a single float2 per lane. B mirrors.
// ---------------------------------------------------------------------------
template <int LDC>
static __device__ __forceinline__ void gemm_tile_body(
    const float* __restrict__ Arow[4], const float* __restrict__ Bmat,
    int ldb, const float* __restrict__ bias,
    float* __restrict__ Cmat, int m0, int n0 /* wave-level origins */)
{
    const int lane = threadIdx.x & 31;
    const int rr   = lane & 15;           // row-within-tile for A/B frags
    const int koff = (lane >> 4) << 1;    // 0 or 2

    const float* Bb[4];
#pragma unroll
    for (int j = 0; j < 4; ++j)
        Bb[j] = Bmat + (size_t)(n0 + j * 16 + rr) * ldb + koff;

    v8f acc[4][4];
#pragma unroll
    for (int j = 0; j < 4; ++j) {
        const float bj = bias[n0 + j * 16 + rr];
#pragma unroll
        for (int i = 0; i < 4; ++i)
#pragma unroll
            for (int e = 0; e < 8; ++e) acc[i][j][e] = bj;
    }

#pragma unroll 2
    for (int k0 = 0; k0 < E_DIM; k0 += 4) {
        v2f af[4], bf[4];
#pragma unroll
        for (int i = 0; i < 4; ++i) af[i] = *(const v2f*)(Arow[i] + k0);
#pragma unroll
        for (int j = 0; j < 4; ++j) bf[j] = *(const v2f*)(Bb[j] + k0);
#pragma unroll
        for (int i = 0; i < 4; ++i)
#pragma unroll
            for (int j = 0; j < 4; ++j)
                acc[i][j] = wmma_f32(af[i], bf[j], acc[i][j]);
    }

    // C layout: VGPR e, lanes 0-15 -> M = e, lanes 16-31 -> M = e + 8
    const int mrow = (lane >> 4) << 3;    // 0 or 8
    const int ncol = lane & 15;
#pragma unroll
    for (int i = 0; i < 4; ++i) {
#pragma unroll
        for (int e = 0; e < 8; ++e) {
            const size_t m = (size_t)(m0 + i * 16 + e + mrow);
#pragma unroll
            for (int j = 0; j < 4; ++j)
                Cmat[m * LDC + n0 + j * 16 + ncol] = acc[i][j][e];
        }
    }
}

// Phase 1: Zx = gather(embed, tokens) @ Wx.T + b    (N = E, ldb = 2E)
__global__ void __launch_bounds__(256)
zx_gemm_kernel(const int* __restrict__ tokens, const float* __restrict__ embed,
               const float* __restrict__ W, const float* __restrict__ bvec,
               float* __restrict__ Zx)
{
    const int wave = threadIdx.x >> 5;
    const int lane = threadIdx.x & 31;
    const int m0 = blockIdx.x * 128 + (wave >> 2) * 64;
    const int n0 = blockIdx.y * 256 + (wave & 3) * 64;
    const int rr   = lane & 15;
    const int koff = (lane >> 4) << 1;

    const float* Arow[4];
#pragma unroll
    for (int i = 0; i < 4; ++i) {
        const int tok = tokens[m0 + i * 16 + rr];
        Arow[i] = embed + (size_t)tok * E_DIM + koff;
    }

    gemm_tile_body<E_DIM>(Arow, W, 2 * E_DIM, bvec, Zx, m0, n0);
}

// Phase 3: logits = Hout @ embed.T + b_cls          (N = V, ldb = E)
__global__ void __launch_bounds__(256)
logits_gemm_kernel(const float* __restrict__ Hout, const float* __restrict__ embed,
                   const float* __restrict__ bcls, float* __restrict__ logits)
{
    const int wave = threadIdx.x >> 5;
    const int lane = threadIdx.x & 31;
    const int m0 = blockIdx.x * 128 + (wave >> 2) * 64;
    const int n0 = blockIdx.y * 256 + (wave & 3) * 64;
    const int rr   = lane & 15;
    const int koff = (lane >> 4) << 1;

    const float* Arow[4];
#pragma unroll
    for (int i = 0; i < 4; ++i)
        Arow[i] = Hout + (size_t)(m0 + i * 16 + rr) * E_DIM + koff;

    gemm_tile_body<V_SZ>(Arow, embed, E_DIM, bcls, logits, m0, n0);
}

// ---------------------------------------------------------------------------
// Phase 2: per-batch sequential scan. 1 WG (1024 thr = 32 waves) per batch.
//   z[j] = Zx[b,t,j] + sum_k h[k] * W[j][1024+k]
//   LayerNorm over E, exact gelu, tanh, cell/hidden update.
// Wave w handles rows j = 32w..32w+31; lanes split K, shfl_xor reduce.
// ---------------------------------------------------------------------------
__global__ void __launch_bounds__(1024)
lstm_scan_kernel(const float* __restrict__ Zx, const float* __restrict__ W,
                 const float* __restrict__ hidden, const float* __restrict__ cell,
                 const float* __restrict__ gamma, const float* __restrict__ beta,
                 float* __restrict__ Hout)
{
    const int b    = blockIdx.x;
    const int tid  = threadIdx.x;
    const int lane = tid & 31;
    const int wave = tid >> 5;

    __shared__ __align__(16) float h_s[E_DIM];
    __shared__ float z_s[E_DIM];
    __shared__ float red_s[32];
    __shared__ float red_q[32];
    __shared__ float s_mu, s_rstd;

    h_s[tid]  = hidden[b * E_DIM + tid];
    float c   = cell[b * E_DIM + tid];
    const float gam = gamma[tid];
    const float bet = beta[tid];
    __syncthreads();

    const float* Wh = W + E_DIM;   // Wh[j][k] = W[j*2048 + 1024 + k]

    for (int t = 0; t < S_LEN; ++t) {
        // ---- GEMV: z = h @ Wh.T (wave-per-row, lane-split K) ----
        for (int r = 0; r < 32; ++r) {
            const int j = (wave << 5) + r;
            const float4* wrow = (const float4*)(Wh + (size_t)j * (2 * E_DIM));
            float acc = 0.f;
#pragma unroll
            for (int kk = 0; kk < 8; ++kk) {
                const float4 wv = wrow[kk * 32 + lane];
                const float4 hv = *(const float4*)(h_s + (kk * 32 + lane) * 4);
                acc += wv.x * hv.x + wv.y * hv.y + wv.z * hv.z + wv.w * hv.w;
            }
#pragma unroll
            for (int off = 16; off; off >>= 1) acc += __shfl_xor(acc, off, 32);
            if (lane == r) z_s[j] = acc;
        }
        __syncthreads();

        const float zj = z_s[tid] + Zx[((size_t)b * S_LEN + t) * E_DIM + tid];

        // ---- LayerNorm statistics (block reduction over 1024) ----
        float s1 = zj, s2 = zj * zj;
#pragma unroll
        for (int off = 16; off; off >>= 1) {
            s1 += __shfl_xor(s1, off, 32);
            s2 += __shfl_xor(s2, off, 32);
        }
        if (lane == 0) { red_s[wave] = s1; red_q[wave] = s2; }
        __syncthreads();
        if (wave == 0) {
            float a = red_s[lane], q = red_q[lane];
#pragma unroll
            for (int off = 16; off; off >>= 1) {
                a += __shfl_xor(a, off, 32);
                q += __shfl_xor(q, off, 32);
            }
            if (lane == 0) {
                const float mu  = a * (1.f / E_DIM);
                const float var = q * (1.f / E_DIM) - mu * mu;
                s_mu   = mu;
                s_rstd = rsqrtf(var + 1e-5f);
            }
        }
        __syncthreads();

        // ---- elementwise: LN -> gelu(exact) / tanh -> cell update ----
        const float n  = (zj - s_mu) * s_rstd * gam + bet;
        const float g  = 0.5f * n * (1.f + erff(n * 0.70710678118654752f));
        const float th = tanhf(n);
        const float cn = g * (c + th);
        const float hn = g * cn;
        c = cn;
        h_s[tid] = hn;
        Hout[((size_t)b * S_LEN + t) * E_DIM + tid] = hn;
        __syncthreads();
    }
}

extern "C" void kernel_launch(void* const* d_in, const int* in_sizes, int n_in,
                              void* d_out, int out_size, void* d_ws, size_t ws_size,
                              hipStream_t stream) {
    const int*   tokens = (const int*)d_in[0];
    const float* hidden = (const float*)d_in[1];
    const float* cell   = (const float*)d_in[2];
    const float* embed  = (const float*)d_in[3];
    const float* W      = (const float*)d_in[4];
    const float* bvec   = (const float*)d_in[5];
    const float* gamma  = (const float*)d_in[6];
    const float* beta   = (const float*)d_in[7];
    const float* bcls   = (const float*)d_in[8];
    float* logits = (float*)d_out;

    float* Zx   = (float*)d_ws;                      // 2048*1024 f32 = 8 MB
    float* Hout = Zx + (size_t)M_TOT * E_DIM;        // 2048*1024 f32 = 8 MB

    // Phase 1: Zx = embed[tokens] @ Wx.T + b   (M=2048, N=1024, K=1024)
    zx_gemm_kernel<<<dim3(M_TOT / 128, E_DIM / 256), 256, 0, stream>>>(
        tokens, embed, W, bvec, Zx);

    // Phase 2: sequential scan, one WG per batch element
    lstm_scan_kernel<<<B_SZ, 1024, 0, stream>>>(
        Zx, W, hidden, cell, gamma, beta, Hout);

    // Phase 3: logits = Hout @ embed.T + b_cls (M=2048, N=32000, K=1024)
    logits_gemm_kernel<<<dim3(M_TOT / 128, V_SZ / 256), 256, 0, stream>>>(
        Hout, embed, bcls, logits);
}